// EdgeConvBlock_43035572306080
// MI455X (gfx1250) — compile-verified
//
#include <hip/hip_runtime.h>

// EdgeConv block for MI455X (gfx1250), wave32, WMMA f32 16x16x4 +
// double-buffered GLOBAL_LOAD_ASYNC_TO_LDS_B128 (ASYNCcnt) for kNN tiles.
//
// Pipeline:
//   k_transpose : x(B,C,N) -> pts(B,N,C) + sq norms
//   k_knn       : async-LDS candidate tiles -> WMMA Gram tiles -> distances
//                 -> per-row top-20 (register insertion sort)
//   k_stats     : edge-feature GEMM via WMMA, per-channel sum/sumsq
//   k_finalize  : fold BN into scale/shift
//   k_out       : recompute GEMM via WMMA, BN+ReLU, max over K, write output

typedef __attribute__((ext_vector_type(2))) float v2f;
typedef __attribute__((ext_vector_type(4))) float f4;
typedef __attribute__((ext_vector_type(8))) float v8f;

#define WMMA_F32(a, b, c) \
  __builtin_amdgcn_wmma_f32_16x16x4_f32(false, (a), false, (b), (short)0, (c), false, false)

constexpr int BATCH = 8;
constexpr int NP    = 2048;
constexpr int CIN   = 64;
constexpr int COUT  = 128;
constexpr int KNN   = 20;
constexpr int CE    = 2 * CIN;  // 128 edge-feature channels
constexpr int QROWS = 32;       // query rows per workgroup in k_knn
constexpr int CHUNK = 64;       // candidate points per chunk in k_knn
constexpr float EPSV = 1e-5f;

// ---- CDNA5 async global->LDS copy (ASYNCcnt), inline asm (portable across
// ROCm7.2 / amdgpu-toolchain since it bypasses the clang builtins).
// Copies 16 bytes per lane; lds_off is the wave-relative LDS byte address
// (low 32 bits of a generic pointer to __shared__), gaddr a 64-bit address.
__device__ __forceinline__ void async_cp16(unsigned int lds_off, const void* g) {
  asm volatile("global_load_async_to_lds_b128 %0, %1, off"
               :: "v"(lds_off), "v"((unsigned long long)(uintptr_t)g)
               : "memory");
}
#define WAIT_ASYNC(n) asm volatile("s_wait_asynccnt %0" :: "n"(n) : "memory")

// ---------------------------------------------------------------- transpose
__global__ __launch_bounds__(256) void k_transpose(const float* __restrict__ x,
                                                   float* __restrict__ pts,
                                                   float* __restrict__ sq) {
  int b = blockIdx.y;
  int n = blockIdx.x * 256 + threadIdx.x;
  const float* xb = x + (size_t)b * CIN * NP;
  float* pr = pts + ((size_t)b * NP + n) * CIN;
  float acc = 0.f;
#pragma unroll 4
  for (int c = 0; c < CIN; ++c) {
    float v = xb[c * NP + n];  // lanes read consecutive n: coalesced
    pr[c] = v;
    acc += v * v;
  }
  sq[b * NP + n] = acc;
}

// ---------------------------------------------------------------- kNN
// One WG (8 wave32) per 32 query points. 64-point candidate chunks are
// double-buffered in LDS via async global->LDS b128 copies; each wave
// computes one 16x16 Gram tile (2 row-tiles x 4 col-tiles) per chunk with
// 16 fp32 WMMAs, overlapped with the next chunk's async copy.
__global__ __launch_bounds__(256) void k_knn(const float* __restrict__ pts,
                                             const float* __restrict__ sq,
                                             int* __restrict__ nn) {
  __shared__ __align__(16) float lA[QROWS * CIN];          // 8 KB query tile
  __shared__ __align__(16) float lCand[2][CHUNK * CIN];    // 2 x 16 KB
  __shared__ float lD[QROWS * CHUNK];                      // 8 KB dist chunk
  __shared__ float lSqQ[QROWS];
  __shared__ float lSqC[CHUNK];

  int b = blockIdx.y;
  int n0 = blockIdx.x * QROWS;
  int t = threadIdx.x;
  int lane = t & 31, wid = t >> 5;
  const float* ptsb = pts + (size_t)b * NP * CIN;

  // query tile: 32x64 = 512 float4, two per thread (coalesced)
  *(f4*)(lA + t * 4)        = *(const f4*)(ptsb + (size_t)n0 * CIN + t * 4);
  *(f4*)(lA + 1024 + t * 4) = *(const f4*)(ptsb + (size_t)n0 * CIN + 1024 + t * 4);
  if (t < QROWS) lSqQ[t] = sq[b * NP + n0 + t];

  // merge lanes: lanes {0,8,16,24} of each wave each own one query row
  bool isMerge = ((lane & 7) == 0);
  int myRow = wid * 4 + (lane >> 3);
  float td[KNN];
  int ti[KNN];
#pragma unroll
  for (int j = 0; j < KNN; ++j) { td[j] = 3.0e38f; ti[j] = 0; }

  int m = lane & 15;
  int koff = (lane < 16) ? 0 : 2;
  int rT = wid >> 2;            // row tile 0..1
  int cT = wid & 3;             // col tile 0..3
  int coll = cT * 16 + m;       // column within chunk this lane owns

  // prologue: async-copy chunk 0 into buffer 0 (16 KB = 4 b128 per thread)
  unsigned int lc0 = (unsigned int)(uintptr_t)&lCand[0][0];
  unsigned int lc1 = (unsigned int)(uintptr_t)&lCand[1][0];
#pragma unroll
  for (int i = 0; i < 4; ++i) {
    int byteOff = (t + i * 256) * 16;
    async_cp16(lc0 + byteOff, (const char*)ptsb + byteOff);
  }

  constexpr int NCH = NP / CHUNK;  // 32 chunks
  for (int c = 0; c < NCH; ++c) {
    int buf = c & 1;
    // issue next chunk's async copy into the other buffer, then wait for
    // this chunk's copies (ASYNCcnt is in-order per wave).
    if (c + 1 < NCH) {
      unsigned int ldst = (c & 1) ? lc0 : lc1;
      const char* gsrc = (const char*)(ptsb + (size_t)(c + 1) * CHUNK * CIN);
#pragma unroll
      for (int i = 0; i < 4; ++i) {
        int byteOff = (t + i * 256) * 16;
        async_cp16(ldst + byteOff, gsrc + byteOff);
      }
      WAIT_ASYNC(4);
    } else {
      WAIT_ASYNC(0);
    }
    if (t < CHUNK) lSqC[t] = sq[b * NP + c * CHUNK + t];
    __syncthreads();  // chunk c resident in lCand[buf] for all waves

    // Gram tile: G = Pq * Pc^T, K=64 contracted in 16 steps of 4
    const float* cand = lCand[buf];
    v8f acc = {};
#pragma unroll
    for (int kk = 0; kk < 16; ++kk) {
      v2f a, bb;
      a.x = lA[(rT * 16 + m) * CIN + kk * 4 + koff];
      a.y = lA[(rT * 16 + m) * CIN + kk * 4 + koff + 1];
      bb.x = cand[coll * CIN + kk * 4 + koff];
      bb.y = cand[coll * CIN + kk * 4 + koff + 1];
      acc = WMMA_F32(a, bb, acc);
    }

    // dist = |q|^2 + |c|^2 - 2 q.c ; write chunk to LDS
    float sqc = lSqC[coll];
#pragma unroll
    for (int v = 0; v < 8; ++v) {
      int row = rT * 16 + v + ((lane < 16) ? 0 : 8);
      lD[row * CHUNK + coll] = lSqQ[row] + sqc - 2.0f * acc[v];
    }
    __syncthreads();

    // merge 64 candidates into sorted top-20 (register insertion sort)
    if (isMerge) {
      const float* dr = lD + myRow * CHUNK;
      int cbase = c * CHUNK;
      for (int j = 0; j < CHUNK; ++j) {
        float d = dr[j];
        if (d < td[KNN - 1]) {
          td[KNN - 1] = d;
          ti[KNN - 1] = cbase + j;
#pragma unroll
          for (int q = KNN - 1; q > 0; --q) {
            if (td[q] < td[q - 1]) {
              float tf = td[q]; td[q] = td[q - 1]; td[q - 1] = tf;
              int tq = ti[q];  ti[q] = ti[q - 1]; ti[q - 1] = tq;
            }
          }
        }
      }
    }
    __syncthreads();  // lD / lSqC / buffers safe for next iteration
  }

  if (isMerge) {
    int* dst = nn + ((size_t)(b * NP + n0 + myRow)) * KNN;
#pragma unroll
    for (int j = 0; j < KNN; ++j) dst[j] = ti[j];
  }
}

// ---------------------------------------------------------------- BN stats
// Edge GEMM pass 1: y = edge(16x128) x W^T(128x128) per row tile; per-channel
// sum / sumsq via shuffle + atomics. W fragments preloaded in registers.
__global__ __launch_bounds__(256) void k_stats(const float* __restrict__ pts,
                                               const int* __restrict__ nn,
                                               const float* __restrict__ W,
                                               const float* __restrict__ bias,
                                               float* __restrict__ sums,
                                               float* __restrict__ sumsq) {
  __shared__ float lE[16 * CE];  // 8 KB edge tile
  __shared__ int lOffC[16], lOffN[16];

  int t = threadIdx.x, lane = t & 31, wid = t >> 5;
  int m = lane & 15;
  int o = wid * 16 + m;                 // output channel this lane owns
  int koff = (lane < 16) ? 0 : 2;

  v2f bw[32];                            // B fragments: W^T[k][o]
#pragma unroll
  for (int kk = 0; kk < 32; ++kk) {
    bw[kk].x = W[o * CE + kk * 4 + koff];
    bw[kk].y = W[o * CE + kk * 4 + koff + 1];
  }
  float bo = bias[o];
  float s = 0.f, s2 = 0.f;

  for (int it = 0; it < 8; ++it) {
    int base = (blockIdx.x * 8 + it) * 16;  // global row base (< 327680)
    __syncthreads();
    if (t < 16) {
      int gr = base + t;
      int bb = gr / (NP * KNN);
      int rem = gr - bb * (NP * KNN);
      int n = rem / KNN;
      int k = rem - n * KNN;
      int nbr = nn[(bb * NP + n) * KNN + k];
      lOffC[t] = (bb * NP + n) * CIN;
      lOffN[t] = (bb * NP + nbr) * CIN;
    }
    __syncthreads();
#pragma unroll
    for (int i = 0; i < 8; ++i) {
      int e = t + i * 256;
      int lr = e >> 7, c = e & 127;
      float v;
      if (c < CIN) v = pts[lOffC[lr] + c];
      else v = pts[lOffN[lr] + c - CIN] - pts[lOffC[lr] + c - CIN];
      lE[lr * CE + c] = v;
    }
    __syncthreads();

    v8f acc = {};
#pragma unroll
    for (int kk = 0; kk < 32; ++kk) {
      v2f a;
      a.x = lE[m * CE + kk * 4 + koff];
      a.y = lE[m * CE + kk * 4 + koff + 1];
      acc = WMMA_F32(a, bw[kk], acc);
    }
#pragma unroll
    for (int v = 0; v < 8; ++v) {
      float y = acc[v] + bo;
      s += y;
      s2 += y * y;
    }
  }
  // lanes l and l+16 hold the same channel o over disjoint rows
  s  += __shfl_down(s, 16, 32);
  s2 += __shfl_down(s2, 16, 32);
  if (lane < 16) {
    atomicAdd(&sums[o], s);
    atomicAdd(&sumsq[o], s2);
  }
}

// ---------------------------------------------------------------- finalize
__global__ void k_finalize(const float* __restrict__ sums,
                           const float* __restrict__ sumsq,
                           const float* __restrict__ gamma,
                           const float* __restrict__ beta,
                           float* __restrict__ scale,
                           float* __restrict__ shift) {
  int o = threadIdx.x;
  float cnt = (float)(BATCH * NP * KNN);
  float mean = sums[o] / cnt;
  float var = sumsq[o] / cnt - mean * mean;
  float sc = gamma[o] * rsqrtf(var + EPSV);
  scale[o] = sc;
  shift[o] = beta[o] - mean * sc;
}

// ---------------------------------------------------------------- output
// Edge GEMM pass 2 per (batch, 16-point group): BN+ReLU in registers,
// max over K via LDS int atomicMax (bits of non-negative floats are ordered).
__global__ __launch_bounds__(256) void k_out(const float* __restrict__ pts,
                                             const int* __restrict__ nn,
                                             const float* __restrict__ W,
                                             const float* __restrict__ bias,
                                             const float* __restrict__ scale,
                                             const float* __restrict__ shift,
                                             float* __restrict__ out) {
  __shared__ float lE[16 * CE];
  __shared__ int lOffC[16], lOffN[16];
  __shared__ int lMax[16 * COUT];  // 8 KB running max (float bits)

  int b = blockIdx.y;
  int n0 = blockIdx.x * 16;
  int t = threadIdx.x, lane = t & 31, wid = t >> 5;
  int m = lane & 15;
  int o = wid * 16 + m;
  int koff = (lane < 16) ? 0 : 2;

  v2f bw[32];
#pragma unroll
  for (int kk = 0; kk < 32; ++kk) {
    bw[kk].x = W[o * CE + kk * 4 + koff];
    bw[kk].y = W[o * CE + kk * 4 + koff + 1];
  }
  float bo = bias[o], sc = scale[o], sh = shift[o];
#pragma unroll
  for (int i = 0; i < 8; ++i) lMax[t + i * 256] = 0;  // relu output >= 0

  for (int rt = 0; rt < 20; ++rt) {  // 20 row tiles = 16 points x 20 neighbors
    __syncthreads();
    if (t < 16) {
      int row = rt * 16 + t;  // 0..319
      int nl = row / KNN;
      int k = row - nl * KNN;
      int n = n0 + nl;
      int nbr = nn[(b * NP + n) * KNN + k];
      lOffC[t] = (b * NP + n) * CIN;
      lOffN[t] = (b * NP + nbr) * CIN;
    }
    __syncthreads();
#pragma unroll
    for (int i = 0; i < 8; ++i) {
      int e = t + i * 256;
      int lr = e >> 7, c = e & 127;
      float v;
      if (c < CIN) v = pts[lOffC[lr] + c];
      else v = pts[lOffN[lr] + c - CIN] - pts[lOffC[lr] + c - CIN];
      lE[lr * CE + c] = v;
    }
    __syncthreads();

    v8f acc = {};
#pragma unroll
    for (int kk = 0; kk < 32; ++kk) {
      v2f a;
      a.x = lE[m * CE + kk * 4 + koff];
      a.y = lE[m * CE + kk * 4 + koff + 1];
      acc = WMMA_F32(a, bw[kk], acc);
    }
#pragma unroll
    for (int v = 0; v < 8; ++v) {
      int row = rt * 16 + v + ((lane < 16) ? 0 : 8);
      int nl = row / KNN;
      float y = (acc[v] + bo) * sc + sh;
      y = fmaxf(y, 0.f);
      atomicMax(&lMax[nl * COUT + o], __float_as_int(y));
    }
  }
  __syncthreads();
  // out[b][o][n] : lanes sweep n for coalescing
#pragma unroll
  for (int i = 0; i < 8; ++i) {
    int e = t + i * 256;
    int nl = e & 15, oo = e >> 4;
    out[((size_t)b * COUT + oo) * NP + n0 + nl] = __int_as_float(lMax[nl * COUT + oo]);
  }
}

// ---------------------------------------------------------------- launch
extern "C" void kernel_launch(void* const* d_in, const int* in_sizes, int n_in,
                              void* d_out, int out_size, void* d_ws, size_t ws_size,
                              hipStream_t stream) {
  const float* x     = (const float*)d_in[0];
  const float* W     = (const float*)d_in[1];
  const float* bias  = (const float*)d_in[2];
  const float* gamma = (const float*)d_in[3];
  const float* beta  = (const float*)d_in[4];
  float* out = (float*)d_out;

  // workspace layout (~5.6 MB)
  float* ws    = (float*)d_ws;
  float* pts   = ws;                              // B*N*CIN
  float* sq    = pts + (size_t)BATCH * NP * CIN;  // B*N
  float* sums  = sq + BATCH * NP;                 // COUT
  float* sumsq = sums + COUT;                     // COUT
  float* scale = sumsq + COUT;                    // COUT
  float* shift = scale + COUT;                    // COUT
  int*   nn    = (int*)(shift + COUT);            // B*N*KNN

  hipMemsetAsync(sums, 0, 2 * COUT * sizeof(float), stream);

  k_transpose<<<dim3(NP / 256, BATCH), 256, 0, stream>>>(x, pts, sq);
  k_knn<<<dim3(NP / QROWS, BATCH), 256, 0, stream>>>(pts, sq, nn);
  k_stats<<<dim3(2560), 256, 0, stream>>>(pts, nn, W, bias, sums, sumsq);
  k_finalize<<<1, COUT, 0, stream>>>(sums, sumsq, gamma, beta, scale, shift);
  k_out<<<dim3(NP / 16, BATCH), 256, 0, stream>>>(pts, nn, W, bias, scale, shift, out);
}